// LocalGINELayer_9921374453964
// MI455X (gfx1250) — compile-verified
//
#include <hip/hip_runtime.h>

// ---------------------------------------------------------------------------
// GINE layer for MI455X (gfx1250, wave32).
// GEMMs run on V_WMMA_F32_16X16X4_F32 (fp32 in / fp32 acc) to match the
// fp32 reference while using the matrix pipe: the kernel is ~memory-bound
// (~650MB HBM -> ~28us floor at 23.3TB/s), so the 33 GFLOP must run well
// above VALU fp32 rates, i.e. through WMMA.
// ---------------------------------------------------------------------------

typedef float v2f __attribute__((ext_vector_type(2)));
typedef float v8f __attribute__((ext_vector_type(8)));

#define HDIM 128
#define EDIM 64
#define BN_EPS 1e-5f

// LDS strides in floats.
// WSTR=136: B-fragment reads hit banks (8k + n) with lane-halves offset by
// 2*136 mod 64 = 16 -> the two 16-lane groups use disjoint bank sets.
#define WSTR 136
// ASTR=68: A reads hit banks 4r(+2 for upper half) -> 32 distinct banks.
#define ASTR 68

__device__ __forceinline__ v8f wmma_f32(v2f a, v2f b, v8f c) {
  // (neg_a, A, neg_b, B, c_mod, C, reuse_a, reuse_b)
  return __builtin_amdgcn_wmma_f32_16x16x4_f32(false, a, false, b, (short)0, c,
                                               false, false);
}

// ---------------------------------------------------------------------------
// Kernel 0: zero agg[N*H] and bn accumulators[2*H] (must re-zero every call).
// ---------------------------------------------------------------------------
__global__ void gine_zero_kernel(float* __restrict__ agg,
                                 float* __restrict__ bnacc, size_t nAgg) {
  const size_t stride = (size_t)gridDim.x * blockDim.x;
  for (size_t i = (size_t)blockIdx.x * blockDim.x + threadIdx.x; i < nAgg;
       i += stride) {
    agg[i] = 0.0f;
  }
  if (blockIdx.x == 0 && threadIdx.x < 2 * HDIM) bnacc[threadIdx.x] = 0.0f;
}

// ---------------------------------------------------------------------------
// Kernel 1: edge GEMM + message + scatter.
// Block = 4 waves. Each wave owns a tile of 16 edges, computes
// relu(ea@W_edge + b_edge + h[src]) for its 16x128 tile via WMMA, then
// atomicAdds into agg[dst]. W_edge stays resident in LDS per block.
// ---------------------------------------------------------------------------
__global__ void __launch_bounds__(128)
gine_edge_kernel(const float* __restrict__ h, const int* __restrict__ ei,
                 const float* __restrict__ ea, const float* __restrict__ W_edge,
                 const float* __restrict__ b_edge, float* __restrict__ agg,
                 int Nn, int Ee) {
  __shared__ __align__(16) float sW[EDIM * WSTR];   // 34816 B
  __shared__ __align__(16) float sA[4][16 * ASTR];  // 17408 B
  const int tid = threadIdx.x;
  const int lane = tid & 31;
  const int wave = tid >> 5;
  const int arow = lane & 15;                 // A row / B col / C col-in-tile
  const int koff = (lane >= 16) ? 2 : 0;      // K sub-offset for upper lanes
  const int mbase = (lane >= 16) ? 8 : 0;     // C row base for upper lanes
  const int* __restrict__ srcI = ei;
  const int* __restrict__ dstI = ei + Ee;

  // Stage W_edge [64 x 128] once per block (L2-resident, cheap to refill).
  for (int i = tid; i < EDIM * HDIM; i += 128)
    sW[(i >> 7) * WSTR + (i & 127)] = W_edge[i];
  __syncthreads();

  const int numWT = Ee >> 4;  // 16-edge wave tiles (E divisible by 16)
  const int numBT = (numWT + 3) >> 2;
  float* sAw = sA[wave];

  for (int bt = blockIdx.x; bt < numBT; bt += gridDim.x) {
    const int wt = bt * 4 + wave;
    if (wt >= numWT) continue;  // wave-uniform: EXEC stays all-ones inside
    const int e0 = wt << 4;

    // Stage A = ea[e0 .. e0+15][0..63] into this wave's padded LDS tile.
    const float4* ea4 = (const float4*)(ea + (size_t)e0 * EDIM);
    for (int i = lane; i < 16 * (EDIM / 4); i += 32) {
      const int r = i >> 4;
      const int q = i & 15;
      ((float4*)&sAw[r * ASTR])[q] = ea4[r * 16 + q];
    }
    // Wave-private tile: same-wave DS ops are in-order; wait + compiler fence.
    asm volatile("s_wait_dscnt 0" ::: "memory");

    int se[8], de[8];
#pragma unroll
    for (int r = 0; r < 8; ++r) {
      se[r] = srcI[e0 + mbase + r];
      de[r] = dstI[e0 + mbase + r];
    }

    const v8f zc = {};
    v8f c[8];
#pragma unroll
    for (int nt = 0; nt < 8; ++nt) c[nt] = zc;

#pragma unroll
    for (int kk = 0; kk < 16; ++kk) {
      const int k0 = kk * 4 + koff;
      const v2f a = *(const v2f*)&sAw[arow * ASTR + k0];
#pragma unroll
      for (int nt = 0; nt < 8; ++nt) {
        v2f b;
        b.x = sW[k0 * WSTR + nt * 16 + arow];
        b.y = sW[(k0 + 1) * WSTR + nt * 16 + arow];
        c[nt] = wmma_f32(a, b, c[nt]);
      }
    }

    // Epilogue: bias + gather h[src] + relu + scatter-add into agg[dst].
#pragma unroll
    for (int nt = 0; nt < 8; ++nt) {
      const int ncol = nt * 16 + arow;
      const float bias = b_edge[ncol];
#pragma unroll
      for (int r = 0; r < 8; ++r) {
        float m = c[nt][r] + bias + h[(size_t)se[r] * HDIM + ncol];
        m = m > 0.0f ? m : 0.0f;
        atomicAdd(agg + (size_t)de[r] * HDIM + ncol, m);
      }
    }
  }
}

// ---------------------------------------------------------------------------
// Kernel 2: t = relu(((1+eps)*h + agg) @ W1 + b1).
// K=128 processed as two 64-row halves of W1 restaged in LDS.
// ---------------------------------------------------------------------------
__global__ void __launch_bounds__(128)
gine_mlp1_kernel(const float* __restrict__ h, const float* __restrict__ agg,
                 const float* __restrict__ W1, const float* __restrict__ b1,
                 const float* __restrict__ epsp, float* __restrict__ t,
                 int Nn) {
  __shared__ __align__(16) float sW[EDIM * WSTR];
  __shared__ __align__(16) float sA[4][16 * ASTR];
  const int tid = threadIdx.x;
  const int lane = tid & 31;
  const int wave = tid >> 5;
  const int arow = lane & 15;
  const int koff = (lane >= 16) ? 2 : 0;
  const int mbase = (lane >= 16) ? 8 : 0;
  const float onePlusEps = 1.0f + epsp[0];

  const int numWT = Nn >> 4;  // N divisible by 16
  const int numBT = (numWT + 3) >> 2;
  float* sAw = sA[wave];

  for (int bt = blockIdx.x; bt < numBT; bt += gridDim.x) {
    const int wt = bt * 4 + wave;
    const bool active = wt < numWT;
    const int n0 = wt << 4;

    const v8f zc = {};
    v8f c[8];
#pragma unroll
    for (int nt = 0; nt < 8; ++nt) c[nt] = zc;

    for (int half = 0; half < 2; ++half) {
      __syncthreads();
      // Stage W1[half*64 .. +63][0..127]
      for (int i = tid; i < EDIM * HDIM; i += 128) {
        const int k = i >> 7, n = i & 127;
        sW[k * WSTR + n] = W1[(size_t)((half << 6) + k) * HDIM + n];
      }
      if (active) {
        // Stage A = (1+eps)*h + agg for this half of K.
        const float4* h4 = (const float4*)(h + (size_t)n0 * HDIM + (half << 6));
        const float4* a4 =
            (const float4*)(agg + (size_t)n0 * HDIM + (half << 6));
        for (int i = lane; i < 16 * 16; i += 32) {
          const int r = i >> 4, q = i & 15;
          const float4 hv = h4[(size_t)r * 32 + q];  // row stride 128 floats
          const float4 av = a4[(size_t)r * 32 + q];
          float4 z;
          z.x = onePlusEps * hv.x + av.x;
          z.y = onePlusEps * hv.y + av.y;
          z.z = onePlusEps * hv.z + av.z;
          z.w = onePlusEps * hv.w + av.w;
          ((float4*)&sAw[r * ASTR])[q] = z;
        }
      }
      __syncthreads();
      if (active) {
#pragma unroll
        for (int kk = 0; kk < 16; ++kk) {
          const int k0 = kk * 4 + koff;
          const v2f a = *(const v2f*)&sAw[arow * ASTR + k0];
#pragma unroll
          for (int nt = 0; nt < 8; ++nt) {
            v2f b;
            b.x = sW[k0 * WSTR + nt * 16 + arow];
            b.y = sW[(k0 + 1) * WSTR + nt * 16 + arow];
            c[nt] = wmma_f32(a, b, c[nt]);
          }
        }
      }
    }
    if (active) {
#pragma unroll
      for (int nt = 0; nt < 8; ++nt) {
        const int ncol = nt * 16 + arow;
        const float bias = b1[ncol];
#pragma unroll
        for (int r = 0; r < 8; ++r) {
          float v = c[nt][r] + bias;
          v = v > 0.0f ? v : 0.0f;
          t[(size_t)(n0 + mbase + r) * HDIM + ncol] = v;
        }
      }
    }
  }
}

// ---------------------------------------------------------------------------
// Kernel 3: z2 = relu(t @ W2 + b2); accumulate per-column sum / sumsq for BN.
// ---------------------------------------------------------------------------
__global__ void __launch_bounds__(128)
gine_mlp2_kernel(const float* __restrict__ t, const float* __restrict__ W2,
                 const float* __restrict__ b2, float* __restrict__ z2,
                 float* __restrict__ bnacc, int Nn) {
  __shared__ __align__(16) float sW[EDIM * WSTR];
  __shared__ __align__(16) float sA[4][16 * ASTR];
  const int tid = threadIdx.x;
  const int lane = tid & 31;
  const int wave = tid >> 5;
  const int arow = lane & 15;
  const int koff = (lane >= 16) ? 2 : 0;
  const int mbase = (lane >= 16) ? 8 : 0;

  const int numWT = Nn >> 4;
  const int numBT = (numWT + 3) >> 2;
  float* sAw = sA[wave];

  for (int bt = blockIdx.x; bt < numBT; bt += gridDim.x) {
    const int wt = bt * 4 + wave;
    const bool active = wt < numWT;
    const int n0 = wt << 4;

    const v8f zc = {};
    v8f c[8];
#pragma unroll
    for (int nt = 0; nt < 8; ++nt) c[nt] = zc;

    for (int half = 0; half < 2; ++half) {
      __syncthreads();
      for (int i = tid; i < EDIM * HDIM; i += 128) {
        const int k = i >> 7, n = i & 127;
        sW[k * WSTR + n] = W2[(size_t)((half << 6) + k) * HDIM + n];
      }
      if (active) {
        const float4* t4 = (const float4*)(t + (size_t)n0 * HDIM + (half << 6));
        for (int i = lane; i < 16 * 16; i += 32) {
          const int r = i >> 4, q = i & 15;
          ((float4*)&sAw[r * ASTR])[q] = t4[(size_t)r * 32 + q];
        }
      }
      __syncthreads();
      if (active) {
#pragma unroll
        for (int kk = 0; kk < 16; ++kk) {
          const int k0 = kk * 4 + koff;
          const v2f a = *(const v2f*)&sAw[arow * ASTR + k0];
#pragma unroll
          for (int nt = 0; nt < 8; ++nt) {
            v2f b;
            b.x = sW[k0 * WSTR + nt * 16 + arow];
            b.y = sW[(k0 + 1) * WSTR + nt * 16 + arow];
            c[nt] = wmma_f32(a, b, c[nt]);
          }
        }
      }
    }
    if (active) {
#pragma unroll
      for (int nt = 0; nt < 8; ++nt) {
        const int ncol = nt * 16 + arow;
        const float bias = b2[ncol];
        float s = 0.0f, s2 = 0.0f;
#pragma unroll
        for (int r = 0; r < 8; ++r) {
          float v = c[nt][r] + bias;
          v = v > 0.0f ? v : 0.0f;
          z2[(size_t)(n0 + mbase + r) * HDIM + ncol] = v;
          s += v;
          s2 += v * v;
        }
        atomicAdd(&bnacc[ncol], s);
        atomicAdd(&bnacc[HDIM + ncol], s2);
      }
    }
  }
}

// ---------------------------------------------------------------------------
// Kernel 4: BN normalize + residual + valid mask.
// ---------------------------------------------------------------------------
__global__ void gine_final_kernel(const float* __restrict__ h,
                                  const float* __restrict__ z2,
                                  const float* __restrict__ valid,
                                  const float* __restrict__ gamma,
                                  const float* __restrict__ beta,
                                  const float* __restrict__ bnacc,
                                  float* __restrict__ out, int Nn) {
  const float invN = 1.0f / (float)Nn;
  const size_t total = (size_t)Nn * HDIM;
  const size_t stride = (size_t)gridDim.x * blockDim.x;
  for (size_t i = (size_t)blockIdx.x * blockDim.x + threadIdx.x; i < total;
       i += stride) {
    const int col = (int)(i & (HDIM - 1));
    const size_t node = i >> 7;
    const float mean = bnacc[col] * invN;
    const float var = bnacc[HDIM + col] * invN - mean * mean;
    const float inv = rsqrtf(var + BN_EPS);
    const float hn = (z2[i] - mean) * inv * gamma[col] + beta[col];
    out[i] = (h[i] + hn) * valid[node];
  }
}

static inline int imin_host(int a, int b) { return a < b ? a : b; }

extern "C" void kernel_launch(void* const* d_in, const int* in_sizes, int n_in,
                              void* d_out, int out_size, void* d_ws,
                              size_t ws_size, hipStream_t stream) {
  const float* h = (const float*)d_in[0];
  const int* ei = (const int*)d_in[1];
  const float* ea = (const float*)d_in[2];
  const float* valid = (const float*)d_in[3];
  const float* W_edge = (const float*)d_in[4];
  const float* b_edge = (const float*)d_in[5];
  const float* W1 = (const float*)d_in[6];
  const float* b1 = (const float*)d_in[7];
  const float* W2 = (const float*)d_in[8];
  const float* b2 = (const float*)d_in[9];
  const float* eps = (const float*)d_in[10];
  const float* gamma = (const float*)d_in[11];
  const float* beta = (const float*)d_in[12];
  float* out = (float*)d_out;

  const int Nn = in_sizes[0] / HDIM;
  const int Ee = in_sizes[2] / EDIM;

  // ws layout: agg[N*H] (reused as z2) | t[N*H] | bnacc[2*H]
  float* agg = (float*)d_ws;
  float* tbuf = agg + (size_t)Nn * HDIM;
  float* bnacc = tbuf + (size_t)Nn * HDIM;

  gine_zero_kernel<<<2048, 256, 0, stream>>>(agg, bnacc, (size_t)Nn * HDIM);

  const int ebt = (Ee / 16 + 3) / 4;
  gine_edge_kernel<<<imin_host(2560, ebt), 128, 0, stream>>>(
      h, ei, ea, W_edge, b_edge, agg, Nn, Ee);

  const int nbt = (Nn / 16 + 3) / 4;
  gine_mlp1_kernel<<<imin_host(2048, nbt), 128, 0, stream>>>(h, agg, W1, b1,
                                                             eps, tbuf, Nn);
  gine_mlp2_kernel<<<imin_host(2048, nbt), 128, 0, stream>>>(tbuf, W2, b2,
                                                             agg /*z2*/, bnacc,
                                                             Nn);
  gine_final_kernel<<<2048, 256, 0, stream>>>(h, agg, valid, gamma, beta,
                                              bnacc, out, Nn);
}